// Latent_40853728919868
// MI455X (gfx1250) — compile-verified
//
#include <hip/hip_runtime.h>
#include <math.h>

typedef __attribute__((ext_vector_type(2))) float v2f;
typedef __attribute__((ext_vector_type(4))) float v4f;
typedef __attribute__((ext_vector_type(8))) float v8f;

#define KN 8192      // rows of z (reduced axis of the LSE)
#define KM 8192      // rows of e (output columns)
#define KD 512       // feature dim
#define GY 8         // number of n-chunks (grid.y)
#define PARTS (GY*4) // softmax partials per column (4 wave n-rows per block)
#define CHUNK (KN/GY)

// D = A(16x4 f32) * B(4x16 f32) + C(16x16 f32)   ->  v_wmma_f32_16x16x4_f32
__device__ __forceinline__ v8f wmma_f32(v2f a, v2f b, v8f c) {
  return __builtin_amdgcn_wmma_f32_16x16x4_f32(
      /*neg_a=*/false, a, /*neg_b=*/false, b,
      /*c_mod=*/(short)0, c, /*reuse_a=*/false, /*reuse_b=*/false);
}

// ---------------- row squared-norms: one wave per row ----------------
__global__ __launch_bounds__(256) void row_norms_kernel(
    const float* __restrict__ x, float* __restrict__ out, int rows) {
  int gid  = blockIdx.x * blockDim.x + threadIdx.x;
  int wave = gid >> 5;
  int lane = gid & 31;
  if (wave >= rows) return;
  const float* r = x + (size_t)wave * KD;
  float s = 0.f;
#pragma unroll
  for (int i = 0; i < KD / 128; ++i) {
    float4 v = *(const float4*)(r + lane * 4 + i * 128);
    s = fmaf(v.x, v.x, fmaf(v.y, v.y, fmaf(v.z, v.z, fmaf(v.w, v.w, s))));
  }
#pragma unroll
  for (int off = 16; off; off >>= 1) s += __shfl_xor(s, off, 32);
  if (lane == 0) out[wave] = s;
}

// ---------- fused GEMM + online-softmax partials (flash-LSE) ----------
// grid = (KM/128, GY); block = 256 threads = 8 waves (4 n-rows x 2 m-cols).
// Each wave: 2(n) x 4(m) tiles of 16x16 => 32 n-rows x 64 m-cols per slab.
// A/B fragments are fetched as b128 covering K=8; the two halves feed two
// WMMAs with a consistent K-permutation between A and B (dot product is
// invariant under a common permutation of K slots).
__global__ __launch_bounds__(256) void dist_lse_partial_kernel(
    const float* __restrict__ z, const float* __restrict__ e,
    const float* __restrict__ z2, const float* __restrict__ e2,
    const float* __restrict__ log_sigma,
    float* __restrict__ pmax, float* __restrict__ psum) {
  const int tid  = threadIdx.x;
  const int w    = tid >> 5;
  const int lane = tid & 31;
  const int wn = w & 3;        // wave's n-row within block (0..3)
  const int wm = w >> 2;       // wave's m-col within block (0..1)
  const int lo = lane & 15;
  const int hi = lane >> 4;    // lane half: K offset 4*hi (permuted-K trick)

  const float ls       = log_sigma[0];
  const float inv_sig2 = __expf(-2.0f * ls);   // 1/sigma^2
  const float alpha    = -0.5f * inv_sig2;     // s = alpha * dist
  const float m2a      = inv_sig2;             // -2*alpha

  const int mb = blockIdx.x * 128 + wm * 64;

  // B operand pointers: column N = mb+16j+lo, K base offset 4*hi folded in
  const float* er[4];
  float cc[4];
#pragma unroll
  for (int j = 0; j < 4; ++j) {
    er[j] = e + (size_t)(mb + 16 * j + lo) * KD + 4 * hi;
    cc[j] = alpha * e2[mb + 16 * j + lo];
  }

  // running (max, sumexp) per m-subtile; column m lives in lanes lo, lo+16
  float mx[4] = {-INFINITY, -INFINITY, -INFINITY, -INFINITY};
  float sm[4] = {0.f, 0.f, 0.f, 0.f};

  auto update = [&](const v8f& c, int nr, int j) {
    float sv[8];
    float smax = -INFINITY;
#pragma unroll
    for (int i = 0; i < 8; ++i) {
      float rc = alpha * z2[nr + i + 8 * hi];     // row constant (L2-hot)
      float s  = fmaf(m2a, c[i], rc + cc[j]);     // alpha*dist
      sv[i] = s;
      smax  = fmaxf(smax, s);
    }
    float mnew = fmaxf(mx[j], smax);
    float acc  = sm[j] * __expf(mx[j] - mnew);    // 0*exp(-inf)=0 first time
#pragma unroll
    for (int i = 0; i < 8; ++i) acc += __expf(sv[i] - mnew);
    mx[j] = mnew;
    sm[j] = acc;
  };

  const int nchunk = blockIdx.y * CHUNK;
  for (int ni = 0; ni < CHUNK / 128; ++ni) {
    const int nb = nchunk + ni * 128 + wn * 32;
    const float* zr0 = z + (size_t)(nb + lo) * KD + 4 * hi;
    const float* zr1 = zr0 + 16 * (size_t)KD;

    v8f c0[4] = {{}, {}, {}, {}};   // n-subtile 0 x m-subtiles 0..3
    v8f c1[4] = {{}, {}, {}, {}};   // n-subtile 1 x m-subtiles 0..3

#pragma unroll 2
    for (int k = 0; k < KD; k += 8) {
      v4f A0 = *(const v4f*)(zr0 + k);
      v4f A1 = *(const v4f*)(zr1 + k);
      v4f B0 = *(const v4f*)(er[0] + k);
      v4f B1 = *(const v4f*)(er[1] + k);
      v4f B2 = *(const v4f*)(er[2] + k);
      v4f B3 = *(const v4f*)(er[3] + k);

      v2f a0l = __builtin_shufflevector(A0, A0, 0, 1);
      v2f a0h = __builtin_shufflevector(A0, A0, 2, 3);
      v2f a1l = __builtin_shufflevector(A1, A1, 0, 1);
      v2f a1h = __builtin_shufflevector(A1, A1, 2, 3);
      v2f b0l = __builtin_shufflevector(B0, B0, 0, 1);
      v2f b0h = __builtin_shufflevector(B0, B0, 2, 3);
      v2f b1l = __builtin_shufflevector(B1, B1, 0, 1);
      v2f b1h = __builtin_shufflevector(B1, B1, 2, 3);
      v2f b2l = __builtin_shufflevector(B2, B2, 0, 1);
      v2f b2h = __builtin_shufflevector(B2, B2, 2, 3);
      v2f b3l = __builtin_shufflevector(B3, B3, 0, 1);
      v2f b3h = __builtin_shufflevector(B3, B3, 2, 3);

      c0[0] = wmma_f32(a0l, b0l, c0[0]);
      c0[1] = wmma_f32(a0l, b1l, c0[1]);
      c0[2] = wmma_f32(a0l, b2l, c0[2]);
      c0[3] = wmma_f32(a0l, b3l, c0[3]);
      c1[0] = wmma_f32(a1l, b0l, c1[0]);
      c1[1] = wmma_f32(a1l, b1l, c1[1]);
      c1[2] = wmma_f32(a1l, b2l, c1[2]);
      c1[3] = wmma_f32(a1l, b3l, c1[3]);

      c0[0] = wmma_f32(a0h, b0h, c0[0]);
      c0[1] = wmma_f32(a0h, b1h, c0[1]);
      c0[2] = wmma_f32(a0h, b2h, c0[2]);
      c0[3] = wmma_f32(a0h, b3h, c0[3]);
      c1[0] = wmma_f32(a1h, b0h, c1[0]);
      c1[1] = wmma_f32(a1h, b1h, c1[1]);
      c1[2] = wmma_f32(a1h, b2h, c1[2]);
      c1[3] = wmma_f32(a1h, b3h, c1[3]);
    }

#pragma unroll
    for (int j = 0; j < 4; ++j) update(c0[j], nb, j);
#pragma unroll
    for (int j = 0; j < 4; ++j) update(c1[j], nb + 16, j);
  }

  // combine the two lane-halves of each column (wave32 xor-16 shuffle)
  const int p = blockIdx.y * 4 + wn;
#pragma unroll
  for (int j = 0; j < 4; ++j) {
    float omx = __shfl_xor(mx[j], 16, 32), osm = __shfl_xor(sm[j], 16, 32);
    float g = fmaxf(mx[j], omx);
    float s = sm[j] * __expf(mx[j] - g) + osm * __expf(omx - g);
    if (hi == 0) {
      size_t m = (size_t)(mb + 16 * j + lo);
      pmax[m * PARTS + p] = g;
      psum[m * PARTS + p] = s;
    }
  }
}

// ---------------- final LSE merge + mean + constants ----------------
__global__ __launch_bounds__(256) void finalize_kernel(
    const float* __restrict__ pmax, const float* __restrict__ psum,
    const float* __restrict__ log_sigma, float* __restrict__ out) {
  __shared__ double red[256];
  int tid = threadIdx.x;
  double local = 0.0;
  for (int m = tid; m < KM; m += 256) {
    const float* pm = pmax + (size_t)m * PARTS;
    const float* ps = psum + (size_t)m * PARTS;
    float g = -INFINITY;
#pragma unroll
    for (int i = 0; i < PARTS; ++i) g = fmaxf(g, pm[i]);
    float s = 0.f;
#pragma unroll
    for (int i = 0; i < PARTS; ++i) s += ps[i] * __expf(pm[i] - g);
    local += (double)(g + logf(s));
  }
  red[tid] = local;
  __syncthreads();
  for (int off = 128; off; off >>= 1) {
    if (tid < off) red[tid] += red[tid + off];
    __syncthreads();
  }
  if (tid == 0) {
    float ls = log_sigma[0];
    double loss = -red[0] / (double)KM
                + 0.5 * (double)KD * (2.0 * (double)ls - 1.0)
                + log((double)KN);
    out[0] = (float)loss;
  }
}

extern "C" void kernel_launch(void* const* d_in, const int* in_sizes, int n_in,
                              void* d_out, int out_size, void* d_ws, size_t ws_size,
                              hipStream_t stream) {
  (void)in_sizes; (void)n_in; (void)out_size; (void)ws_size;
  const float* z         = (const float*)d_in[0];
  const float* e         = (const float*)d_in[1];
  const float* log_sigma = (const float*)d_in[2];
  float* out = (float*)d_out;

  float* z2   = (float*)d_ws;                 // KN floats
  float* e2   = z2 + KN;                      // KM floats
  float* pmax = e2 + KM;                      // KM*PARTS floats
  float* psum = pmax + (size_t)KM * PARTS;    // KM*PARTS floats (~2.2 MB total)

  row_norms_kernel<<<KN * 32 / 256, 256, 0, stream>>>(z, z2, KN);
  row_norms_kernel<<<KM * 32 / 256, 256, 0, stream>>>(e, e2, KM);

  dim3 grid(KM / 128, GY);
  dist_lse_partial_kernel<<<grid, 256, 0, stream>>>(z, e, z2, e2, log_sigma,
                                                    pmax, psum);

  finalize_kernel<<<1, 256, 0, stream>>>(pmax, psum, log_sigma, out);
}